// PositionFeature_69458211111350
// MI455X (gfx1250) — compile-verified
//
#include <hip/hip_runtime.h>
#include <stdint.h>

typedef __attribute__((ext_vector_type(2))) float        v2f;
typedef __attribute__((ext_vector_type(8))) float        v8f;
typedef __attribute__((ext_vector_type(4))) unsigned int v4u;
typedef __attribute__((ext_vector_type(8))) int          v8i;
typedef __attribute__((ext_vector_type(4))) int          v4i;

#define N_ 16
#define S_ 4
#define C_ 128
#define H_ 36
#define W_ 36
#define U_ 8000
#define R_ 4

#define WAVES_PER_BLOCK 2
#define UPW 8                              // (s,u) tasks per wave (8000 % 8 == 0)
#define WIN_DW (N_ * C_ * 4)               // 8192 dwords: [n][c][y][x] window
#define PAD_DW ((WIN_DW / 256) * 4)        // TDM pad: 4 dwords per 256 -> 128
#define BUF_DW (WIN_DW + PAD_DW)           // 8320 dwords = 33280 B per buffer
#define LDS_BYTES (WAVES_PER_BLOCK * 2 * BUF_DW * 4)   // 133120 B

#if defined(__has_builtin)
#if __has_builtin(__builtin_amdgcn_tensor_load_to_lds)
#define HAVE_TDM 1
#endif
#endif

// Padded LDS dword offset of the 2x2 window of (n, c):
// e = n*512 + c*4; off = e + (e>>8)*4 = n*520 + c*4 + 4*[c>=64]
__device__ __forceinline__ int lds_off_dw(int n, int c) {
    return n * 520 + c * 4 + ((c >= 64) ? 4 : 0);
}

// grid_sample bilinear (align_corners=False, border): clamp window base to
// [0, size-2] and clamp fractional weights to [0,1] -> exact border semantics.
__device__ __forceinline__ void window_params(const float* __restrict__ positions,
                                              int u, int& xb, int& yb, float4& wv) {
    const float fx = ((positions[2 * u + 0] + 1.0f) * (float)W_ - 1.0f) * 0.5f;
    const float fy = ((positions[2 * u + 1] + 1.0f) * (float)H_ - 1.0f) * 0.5f;
    xb = (int)floorf(fx);
    yb = (int)floorf(fy);
    xb = xb < 0 ? 0 : (xb > W_ - 2 ? W_ - 2 : xb);
    yb = yb < 0 ? 0 : (yb > H_ - 2 ? H_ - 2 : yb);
    float wx = fx - (float)xb; wx = wx < 0.0f ? 0.0f : (wx > 1.0f ? 1.0f : wx);
    float wy = fy - (float)yb; wy = wy < 0.0f ? 0.0f : (wy > 1.0f ? 1.0f : wy);
    wv.x = (1.0f - wx) * (1.0f - wy);
    wv.y = wx * (1.0f - wy);
    wv.z = (1.0f - wx) * wy;
    wv.w = wx * wy;
}

// Issue the 4-D TDM tile load: x(2,s=1) y(2,s=36) c(128,s=1296) n(16,s=512*1296)
__device__ __forceinline__ void issue_window_load(const float* __restrict__ core,
                                                  const float* __restrict__ positions,
                                                  int s, int u, float* buf, int lane) {
    int xb, yb; float4 wv;
    window_params(positions, u, xb, yb, wv);
#ifdef HAVE_TDM
    const uint64_t gaddr = (uint64_t)(uintptr_t)core +
        4ull * (uint64_t)((s * C_) * (H_ * W_) + yb * W_ + xb);
    // low 32 bits of a generic LDS pointer == LDS byte offset (ISA 10.2)
    const uint32_t lds = (uint32_t)__builtin_amdgcn_readfirstlane(
        (int)(uint32_t)(uintptr_t)buf);
    const uint32_t alo = (uint32_t)__builtin_amdgcn_readfirstlane((int)(uint32_t)gaddr);
    const uint32_t ahi = (uint32_t)__builtin_amdgcn_readfirstlane((int)(uint32_t)(gaddr >> 32));

    // D# group0: count=1 | lds_addr | global_addr[56:0] | type=2
    v4u g0 = { 1u, lds, alo, (ahi & 0x01FFFFFFu) | (2u << 30) };
    // D# group1: data_size=4B, pad_enable, pad_interval=256dw, pad_amount=4dw;
    // tensor_dim0=2, tensor_dim1=2, tile_dim0=2, tile_dim1=2, tile_dim2=128,
    // dim0_stride=36, dim1_stride=1296
    v8i g1 = { (int)((2u << 16) | (1u << 20) | (7u << 22) | (3u << 25)),
               (int)(2u << 16),            // tensor_dim0[15:0]=2
               (int)(2u << 16),            // tensor_dim1[15:0]=2
               (int)(2u << 16),            // tile_dim0=2
               (int)(2u | (128u << 16)),   // tile_dim1=2, tile_dim2=128
               36,                         // tensor_dim0_stride
               (int)(1296u << 16),         // tensor_dim1_stride[15:0]
               0 };
    // D# group2: tensor_dim2=128, tensor_dim3=16, dim2_stride=512*1296, tile_dim3=16
    v4i g2 = { 128, 16, 663552, (int)(16u << 16) };
    v4i g3 = { 0, 0, 0, 0 };
#if __clang_major__ >= 23
    v8i gx = { 0, 0, 0, 0, 0, 0, 0, 0 };
    __builtin_amdgcn_tensor_load_to_lds(g0, g1, g2, g3, gx, 0);
#else
    __builtin_amdgcn_tensor_load_to_lds(g0, g1, g2, g3, 0);
#endif
#else
    // Fallback: cooperative global->LDS staging with the same padded layout.
    for (int e = lane; e < N_ * C_; e += 32) {
        const int n = e >> 7, c = e & (C_ - 1);
        const float* p = core + (size_t)(n * (S_ * C_) + s * C_ + c) * (H_ * W_)
                              + (size_t)yb * W_ + xb;
        float4 q; q.x = p[0]; q.y = p[1]; q.z = p[W_]; q.w = p[W_ + 1];
        *(float4*)(buf + lds_off_dw(n, c)) = q;
    }
#endif
}

__device__ __forceinline__ void compute_task(const float* __restrict__ feature,
                                             const float* __restrict__ bias,
                                             const float* __restrict__ positions,
                                             float* __restrict__ out,
                                             const float* buf, int s, int u, int lane) {
    int xb, yb; float4 wv;
    window_params(positions, u, xb, yb, wv);

    // V_WMMA_F32_16X16X4_F32: A 16x4 (M=r padded), B 4x16 (N=n batch)
    const int n   = lane & 15;
    const int rA  = (lane & 15) % R_;   // duplicate r for pad rows 4..15
    const int kHi = (lane >> 4) * 2;    // K-slots {kHi,kHi+1}

    // Per-lane bases; all per-step offsets below are compile-time immediates.
    const float* featRow = feature + (((size_t)(s * U_ + u) * R_ + rA) * C_ + kHi);
    const float* bufn    = buf + (n * 520 + kHi * 4);

    // software-pipelined (distance 1), fully unrolled: immediate DS/VMEM offsets
    v2f    a  = *(const v2f*)(featRow);
    float4 q0 = *(const float4*)(bufn + 0);
    float4 q1 = *(const float4*)(bufn + 4);

    v8f acc = {};
    #pragma unroll
    for (int kk = 0; kk < C_ / 4; ++kk) {
        v2f an = a; float4 q0n = q0, q1n = q1;
        if (kk < C_ / 4 - 1) {
            const int off = (kk + 1) * 16 + (((kk + 1) >= 16) ? 4 : 0); // dwords
            an  = *(const v2f*)(featRow + (kk + 1) * 4);
            q0n = *(const float4*)(bufn + off);
            q1n = *(const float4*)(bufn + off + 4);
        }
        v2f b;
        b[0] = fmaf(wv.x, q0.x, fmaf(wv.y, q0.y, fmaf(wv.z, q0.z, wv.w * q0.w)));
        b[1] = fmaf(wv.x, q1.x, fmaf(wv.y, q1.y, fmaf(wv.z, q1.z, wv.w * q1.w)));
        acc = __builtin_amdgcn_wmma_f32_16x16x4_f32(
                  false, a, false, b, (short)0, acc, false, false);
        a = an; q0 = q0n; q1 = q1n;
    }

    if (lane < 16) {   // D rows M=0..3 live in acc[0..3] on lanes 0..15 (N=n)
        const float* bp = bias + ((size_t)(s * U_ + u) * R_);
        float4 o;
        o.x = acc[0] + bp[0];
        o.y = acc[1] + bp[1];
        o.z = acc[2] + bp[2];
        o.w = acc[3] + bp[3];
        *(float4*)(out + (((size_t)(n * S_ + s) * U_ + u) * R_)) = o;
    }
}

__global__ __launch_bounds__(WAVES_PER_BLOCK * 32)
void PositionFeature_tdm_wmma_kernel(const float* __restrict__ core,
                                     const float* __restrict__ positions,
                                     const float* __restrict__ feature,
                                     const float* __restrict__ bias,
                                     float* __restrict__ out) {
    extern __shared__ float smem[];
    const int lane = threadIdx.x & 31;
    const int wave = threadIdx.x >> 5;
    const int g    = blockIdx.x * WAVES_PER_BLOCK + wave;
    const int w0   = g * UPW;            // UPW consecutive (s,u) tasks
    const int s    = w0 / U_;            // uniform per wave (8000 % UPW == 0)
    const int ub   = w0 % U_;

    float* buf0 = smem + wave * (2 * BUF_DW);
    float* buf1 = buf0 + BUF_DW;

    issue_window_load(core, positions, s, ub, buf0, lane);
    for (int i = 0; i < UPW - 1; ++i) {
        float* cur = (i & 1) ? buf1 : buf0;
        float* nxt = (i & 1) ? buf0 : buf1;
        issue_window_load(core, positions, s, ub + i + 1, nxt, lane);
#ifdef HAVE_TDM
        __builtin_amdgcn_s_wait_tensorcnt(1);   // window i landed in LDS
#endif
        compute_task(feature, bias, positions, out, cur, s, ub + i, lane);
    }
#ifdef HAVE_TDM
    __builtin_amdgcn_s_wait_tensorcnt(0);
#endif
    compute_task(feature, bias, positions, out,
                 ((UPW - 1) & 1) ? buf1 : buf0, s, ub + UPW - 1, lane);
}

extern "C" void kernel_launch(void* const* d_in, const int* in_sizes, int n_in,
                              void* d_out, int out_size, void* d_ws, size_t ws_size,
                              hipStream_t stream) {
    const float* core      = (const float*)d_in[0];
    const float* positions = (const float*)d_in[1];
    const float* feature   = (const float*)d_in[2];
    const float* bias      = (const float*)d_in[3];
    float* out             = (float*)d_out;

    (void)hipFuncSetAttribute((const void*)PositionFeature_tdm_wmma_kernel,
                              hipFuncAttributeMaxDynamicSharedMemorySize, LDS_BYTES);

    const int tasks  = S_ * U_;                              // 32000
    const int blocks = tasks / (UPW * WAVES_PER_BLOCK);      // 2000
    PositionFeature_tdm_wmma_kernel<<<blocks, WAVES_PER_BLOCK * 32, LDS_BYTES, stream>>>(
        core, positions, feature, bias, out);
}